// EEGM2_Light_1812476199778
// MI455X (gfx1250) — compile-verified
//
#include <hip/hip_runtime.h>
#include <math.h>

// ---------------------------------------------------------------------------
// EEGM2-Light pipeline for gfx1250 (MI455X).  All GEMM-shaped stages use
// V_WMMA_F32_16X16X4_F32 (fp32 A/B, fp32 accum) so numerics match the f32
// reference; workload (~15 GFLOP over ~150MB, L2-resident) is bandwidth
// bound, so fp32 WMMA is the right precision/throughput point.
// This revision: branch-free boundary masking (no saveexec per load) and
// float2 (b64) loads for all contiguous A/B operand pairs; conv weights are
// repacked [tap][out][ch] so weight pairs are contiguous too.
// ---------------------------------------------------------------------------

typedef __attribute__((ext_vector_type(2))) float v2f;
typedef __attribute__((ext_vector_type(8))) float v8f;

static __device__ __forceinline__ v8f wmma4(float ax, float ay, float bx, float by, v8f c) {
  v2f a; a.x = ax; a.y = ay;
  v2f b; b.x = bx; b.y = by;
  return __builtin_amdgcn_wmma_f32_16x16x4_f32(false, a, false, b, (short)0, c,
                                               false, false);
}
static __device__ __forceinline__ v8f wmma4v(v2f a, v2f b, v8f c) {
  return __builtin_amdgcn_wmma_f32_16x16x4_f32(false, a, false, b, (short)0, c,
                                               false, false);
}

static __device__ __forceinline__ float sigm(float x) { return 1.f / (1.f + __expf(-x)); }

#define Z8 {0.f,0.f,0.f,0.f,0.f,0.f,0.f,0.f}

// ---------------------------------------------------------------------------
// Prep: fold conv1/conv3/conv7 + fuse(1x1) + linear into one k=7 conv.
// ---------------------------------------------------------------------------
__global__ void k_prep_weff(const float* __restrict__ fuse_w,
                            const float* __restrict__ w1,
                            const float* __restrict__ w3,
                            const float* __restrict__ w7,
                            float* __restrict__ Weff) {
  int idx = blockIdx.x * blockDim.x + threadIdx.x;   // 7*64*64
  if (idx >= 7 * 64 * 64) return;
  int tap = idx / 4096;
  int o   = (idx >> 6) & 63;
  int i   = idx & 63;
  float acc = 0.f;
  for (int m = 0; m < 64; ++m)
    acc += fuse_w[o * 192 + 128 + m] * w7[(m * 64 + i) * 7 + tap];
  if (tap >= 2 && tap <= 4)
    for (int m = 0; m < 64; ++m)
      acc += fuse_w[o * 192 + 64 + m] * w3[(m * 64 + i) * 3 + (tap - 2)];
  if (tap == 3)
    for (int m = 0; m < 64; ++m)
      acc += fuse_w[o * 192 + m] * w1[m * 64 + i];
  Weff[(o * 64 + i) * 7 + tap] = acc;
}

// Writes W2 in [tap][d][i] layout so (i, i+1) weight pairs are contiguous.
__global__ void k_prep_w2(const float* __restrict__ lin_w,
                          const float* __restrict__ Weff,
                          float* __restrict__ W2t) {
  int idx = blockIdx.x * blockDim.x + threadIdx.x;
  if (idx >= 7 * 64 * 64) return;
  int tap = idx / 4096;
  int d   = (idx >> 6) & 63;
  int i   = idx & 63;
  float acc = 0.f;
  for (int o = 0; o < 64; ++o)
    acc += lin_w[d * 64 + o] * Weff[(o * 64 + i) * 7 + tap];
  W2t[((size_t)tap * 64 + d) * 64 + i] = acc;
}

__global__ void k_prep_bias(const float* __restrict__ fuse_w, const float* __restrict__ fuse_b,
                            const float* __restrict__ b1, const float* __restrict__ b3,
                            const float* __restrict__ b7,
                            const float* __restrict__ lin_w, const float* __restrict__ lin_b,
                            float* __restrict__ bias2) {
  __shared__ float eb[64];
  int d = threadIdx.x;
  float acc = fuse_b[d];
  for (int m = 0; m < 64; ++m)
    acc += fuse_w[d * 192 + m] * b1[m] + fuse_w[d * 192 + 64 + m] * b3[m] +
           fuse_w[d * 192 + 128 + m] * b7[m];
  eb[d] = acc;
  __syncthreads();
  float r = lin_b[d];
  for (int o = 0; o < 64; ++o) r += lin_w[d * 64 + o] * eb[o];
  bias2[d] = r;
}

// Repack enc2_w (128,64,3) -> [tap][o][i]
__global__ void k_prep_enc2(const float* __restrict__ w, float* __restrict__ wt) {
  int idx = blockIdx.x * blockDim.x + threadIdx.x;  // 128*64*3 = 24576
  if (idx >= 24576) return;
  int tap = idx / 8192;
  int o   = (idx >> 6) & 127;
  int i   = idx & 63;
  wt[((size_t)tap * 128 + o) * 64 + i] = w[(o * 64 + i) * 3 + tap];
}

// Repack enc3_w (256,128,3) -> [tap][o][i]
__global__ void k_prep_enc3(const float* __restrict__ w, float* __restrict__ wt) {
  int idx = blockIdx.x * blockDim.x + threadIdx.x;  // 256*128*3 = 98304
  if (idx >= 98304) return;
  int tap = idx / 32768;
  int o   = (idx >> 7) & 255;
  int i   = idx & 127;
  wt[((size_t)tap * 256 + o) * 128 + i] = w[(o * 128 + i) * 3 + tap];
}

// ---------------------------------------------------------------------------
// K1: fused embed: h[b,t,d] = sum_tap sum_ch W2[tap,d,ch] x[b,ch,t+tap-3]
// ---------------------------------------------------------------------------
__global__ void k_embed(const float* __restrict__ x, const float* __restrict__ W2t,
                        const float* __restrict__ b2, float* __restrict__ hbuf) {
  int m0   = blockIdx.x * 16;
  int bidx = blockIdx.y;
  int wave = threadIdx.x >> 5;
  int lane = threadIdx.x & 31;
  int half = lane >> 4, lr = lane & 15;
  int n0 = wave * 16;
  v8f acc = Z8;
  const float* xb = x + (size_t)bidx * 64 * 2048;
  for (int tap = 0; tap < 7; ++tap) {
    int t = m0 + lr + tap - 3;
    float msk = (t >= 0 && t < 2048) ? 1.f : 0.f;
    int tc = t < 0 ? 0 : (t > 2047 ? 2047 : t);
    const v2f* wp = (const v2f*)(W2t + ((size_t)tap * 64 + n0 + lr) * 64);
    for (int kb = 0; kb < 16; ++kb) {
      int kA = kb * 4 + half * 2;
      float ax = xb[(size_t)kA * 2048 + tc] * msk;
      float ay = xb[(size_t)(kA + 1) * 2048 + tc] * msk;
      v2f bv = wp[kb * 2 + half];
      acc = wmma4(ax, ay, bv.x, bv.y, acc);
    }
  }
  float bb = b2[n0 + lr];
  for (int r = 0; r < 8; ++r) {
    int m = m0 + r + half * 8;
    hbuf[((size_t)bidx * 2048 + m) * 64 + n0 + lr] = acc[r] + bb;
  }
}

// ---------------------------------------------------------------------------
// K2: in_proj (32768 x 64) @ (64 x 384) -> z (128), xBCraw (256)
// ---------------------------------------------------------------------------
__global__ void k_inproj(const float* __restrict__ hbuf, const float* __restrict__ in_w,
                         float* __restrict__ z, float* __restrict__ xBCr) {
  int m0 = blockIdx.x * 16;
  int wave = threadIdx.x >> 5;
  int lane = threadIdx.x & 31;
  int half = lane >> 4, lr = lane & 15;
  int n0 = blockIdx.y * 64 + wave * 16;
  v8f acc = Z8;
  const v2f* ap = (const v2f*)(hbuf + (size_t)(m0 + lr) * 64);
  const v2f* bp = (const v2f*)(in_w + (size_t)(n0 + lr) * 64);
  for (int kb = 0; kb < 16; ++kb) {
    acc = wmma4v(ap[kb * 2 + half], bp[kb * 2 + half], acc);
  }
  int n = n0 + lr;
  for (int r = 0; r < 8; ++r) {
    int m = m0 + r + half * 8;
    float v = acc[r];
    if (n < 128) z[(size_t)m * 128 + n] = v;
    else         xBCr[(size_t)m * 256 + (n - 128)] = v;
  }
}

// K2b: dt = softplus(h . in_w[384+h] + dt_bias)
__global__ void k_dt(const float* __restrict__ hbuf, const float* __restrict__ in_w,
                     const float* __restrict__ dt_bias, float* __restrict__ dtb) {
  int idx = blockIdx.x * blockDim.x + threadIdx.x;  // 65536
  int m = idx >> 1, hh = idx & 1;
  float acc = dt_bias[hh];
  const float* wrow = in_w + (384 + hh) * 64;
  const float* hr = hbuf + (size_t)m * 64;
  for (int k = 0; k < 64; ++k) acc += wrow[k] * hr[k];
  dtb[idx] = (acc > 20.f) ? acc : log1pf(__expf(acc));
}

// K3: causal depthwise conv (k=4) + silu; also xsd = xs * dt
__global__ void k_conv_silu(const float* __restrict__ xBCr, const float* __restrict__ conv_w,
                            const float* __restrict__ conv_b, const float* __restrict__ dtb,
                            float* __restrict__ xBCa, float* __restrict__ xsd) {
  int idx = blockIdx.x * 256 + threadIdx.x;  // B*L*256
  int ch = idx & 255;
  int m  = idx >> 8;
  int t  = m & 2047;
  float acc = conv_b[ch];
  for (int k = 0; k < 4; ++k) {
    int tt = t - 3 + k;
    if (tt >= 0) acc += conv_w[ch * 4 + k] * xBCr[(size_t)(m - 3 + k) * 256 + ch];
  }
  float act = acc * sigm(acc);
  xBCa[(size_t)m * 256 + ch] = act;
  if (ch < 128) {
    int hh = ch >> 6;
    xsd[(size_t)m * 128 + ch] = act * dtb[m * 2 + hh];
  }
}

// K4: per-(b,h,chunk) cumsum of dt*A; G=exp(Acs); ds=exp(AL-Acs); AL per chunk
__global__ void k_scan(const float* __restrict__ dtb, const float* __restrict__ A_log,
                       float* __restrict__ Acs, float* __restrict__ G,
                       float* __restrict__ ds, float* __restrict__ AL) {
  int tid = threadIdx.x;  // 256 = 16b * 2h * 8c
  int b = tid >> 4, hh = (tid >> 3) & 1, c = tid & 7;
  float Aneg = -__expf(A_log[hh]);
  int base  = (b * 2 + hh) * 2048 + c * 256;
  int mbase = b * 2048 + c * 256;
  float acc = 0.f;
  for (int t = 0; t < 256; ++t) {
    acc += dtb[(mbase + t) * 2 + hh] * Aneg;
    Acs[base + t] = acc;
    G[base + t]   = __expf(acc);
  }
  AL[tid] = acc;
  for (int t = 0; t < 256; ++t) ds[base + t] = __expf(acc - Acs[base + t]);
}

// K5: per-(b,c,h) states S[n][p] = sum_t B[t,n] * ds[t] * Xd[t,p]   (WMMA K=256)
__global__ void k_states(const float* __restrict__ xBCa, const float* __restrict__ xsd,
                         const float* __restrict__ ds, float* __restrict__ S) {
  int blk = blockIdx.x;  // 256
  int b = blk >> 4, c = (blk >> 1) & 7, hh = blk & 1;
  int wave = threadIdx.x >> 5, lane = threadIdx.x & 31;
  int half = lane >> 4, lr = lane & 15;
  int n0 = wave * 16;
  size_t t0 = (size_t)b * 2048 + c * 256;
  int dbase = (b * 2 + hh) * 2048 + c * 256;
  for (int ps = 0; ps < 4; ++ps) {
    int p0 = ps * 16;
    v8f acc = Z8;
    for (int kb = 0; kb < 64; ++kb) {
      int kA = kb * 4 + half * 2;
      float ax = xBCa[(t0 + kA) * 256 + 128 + n0 + lr];
      float ay = xBCa[(t0 + kA + 1) * 256 + 128 + n0 + lr];
      float bx = xsd[(t0 + kA) * 128 + hh * 64 + p0 + lr] * ds[dbase + kA];
      float by = xsd[(t0 + kA + 1) * 128 + hh * 64 + p0 + lr] * ds[dbase + kA + 1];
      acc = wmma4(ax, ay, bx, by, acc);
    }
    size_t sb = (((size_t)b * 8 + c) * 2 + hh) * 64;
    for (int r = 0; r < 8; ++r)
      S[(sb + n0 + r + half * 8) * 64 + p0 + lr] = acc[r];
  }
}

// K5b: inter-chunk recurrence Sin[c] = exp(AL[c-1])*Sin[c-1] + S[c-1]
__global__ void k_chunkscan(const float* __restrict__ S, const float* __restrict__ AL,
                            float* __restrict__ Sin) {
  int bh = blockIdx.x;            // 0..31 : b*2+h
  int b = bh >> 1, hh = bh & 1;
  int tid = threadIdx.x;          // 256 threads, 16 elems each
  for (int e = 0; e < 16; ++e) {
    int idx = tid * 16 + e;       // 0..4095
    int n = idx >> 6, p = idx & 63;
    float s = 0.f;
    for (int c = 0; c < 8; ++c) {
      size_t off = ((((size_t)b * 8 + c) * 2 + hh) * 64 + n) * 64 + p;
      Sin[off] = s;
      s = __expf(AL[bh * 8 + c]) * s + S[off];
    }
  }
}

// K6: SSD output.  Per block: (b, chunk, head, 32-row t tile).
// Stage A: M[t,s] = (C.B^T)[t,s] * exp(Acs[t]-Acs[s]) (masked) -> LDS
// Stage B: Ydiag = M . Xd ;  Stage C: Yoff = (C . Sin) * G[t] ; + D*x
__global__ void k_ssd_y(const float* __restrict__ xBCa, const float* __restrict__ xsd,
                        const float* __restrict__ Acs, const float* __restrict__ G,
                        const float* __restrict__ Sin, const float* __restrict__ Dp,
                        float* __restrict__ y) {
  __shared__ float Msh[32 * 260];
  int t32 = blockIdx.x;      // 0..7
  int b   = blockIdx.y;      // 0..15
  int c   = blockIdx.z >> 1; // 0..7
  int hh  = blockIdx.z & 1;
  int wave = threadIdx.x >> 5, lane = threadIdx.x & 31;
  int half = lane >> 4, lr = lane & 15;
  int gbh = (b * 2 + hh) * 2048 + c * 256;
  size_t tok0 = (size_t)b * 2048 + c * 256;
  int tb = t32 * 32;

  {  // stage A: C.B^T with decay mask -> LDS
    int tsub = wave & 1;
    int ssbase = (wave >> 1) * 4;
    int trow = tb + tsub * 16 + lr;
    const v2f* apC = (const v2f*)(xBCa + (tok0 + trow) * 256 + 192);
    for (int j = 0; j < 4; ++j) {
      int ssub = ssbase + j;
      int srow = ssub * 16 + lr;
      const v2f* bpB = (const v2f*)(xBCa + (tok0 + srow) * 256 + 128);
      v8f acc = Z8;
      for (int kb = 0; kb < 16; ++kb) {
        acc = wmma4v(apC[kb * 2 + half], bpB[kb * 2 + half], acc);
      }
      for (int r = 0; r < 8; ++r) {
        int tloc = tsub * 16 + r + half * 8;
        int tch = tb + tloc;
        int sch = ssub * 16 + lr;
        float v = 0.f;
        if (sch <= tch) v = acc[r] * __expf(Acs[gbh + tch] - Acs[gbh + sch]);
        Msh[tloc * 260 + sch] = v;
      }
    }
  }
  __syncthreads();
  {  // stage B + C
    int tsub = wave & 1;
    int psub = wave >> 1;  // 0..3
    int p0 = psub * 16;
    v8f accB = Z8, accC = Z8;
    const v2f* mp = (const v2f*)(Msh + (tsub * 16 + lr) * 260);
    for (int kb = 0; kb < 64; ++kb) {
      int kA = kb * 4 + half * 2;
      v2f av = mp[kb * 2 + half];
      float bx = xsd[(tok0 + kA) * 128 + hh * 64 + p0 + lr];
      float by = xsd[(tok0 + kA + 1) * 128 + hh * 64 + p0 + lr];
      accB = wmma4(av.x, av.y, bx, by, accB);
    }
    size_t sb = (((size_t)b * 8 + c) * 2 + hh) * 64;
    int trowA = tb + tsub * 16 + lr;
    const v2f* apC = (const v2f*)(xBCa + (tok0 + trowA) * 256 + 192);
    for (int kb = 0; kb < 16; ++kb) {
      int kA = kb * 4 + half * 2;
      float bx = Sin[(sb + kA) * 64 + p0 + lr];
      float by = Sin[(sb + kA + 1) * 64 + p0 + lr];
      v2f av = apC[kb * 2 + half];
      accC = wmma4(av.x, av.y, bx, by, accC);
    }
    float dpv = Dp[hh];
    for (int r = 0; r < 8; ++r) {
      int tloc = tsub * 16 + r + half * 8;
      int tch = tb + tloc;
      size_t tok = tok0 + tch;
      int p = p0 + lr;
      float xv = xBCa[tok * 256 + hh * 64 + p];
      y[tok * 128 + hh * 64 + p] = accB[r] + G[gbh + tch] * accC[r] + dpv * xv;
    }
  }
}

// K7a: u = y * silu(z); RMS-normalize over 128; write normalized u back into z
__global__ void k_gate_rms(const float* __restrict__ y, float* __restrict__ z,
                           const float* __restrict__ norm_w) {
  __shared__ float red[128];
  int tok = blockIdx.x;
  int d = threadIdx.x;
  size_t idx = (size_t)tok * 128 + d;
  float zv = z[idx];
  float u = y[idx] * zv * sigm(zv);
  red[d] = u * u;
  __syncthreads();
  for (int s = 64; s > 0; s >>= 1) { if (d < s) red[d] += red[d + s]; __syncthreads(); }
  float rms = rsqrtf(red[0] * (1.f / 128.f) + 1e-5f);
  z[idx] = u * rms * norm_w[d];
}

// K7b: out_proj (32768 x 128) @ (128 x 64)
__global__ void k_outproj(const float* __restrict__ u, const float* __restrict__ out_w,
                          float* __restrict__ mbuf) {
  int m0 = blockIdx.x * 16;
  int wave = threadIdx.x >> 5, lane = threadIdx.x & 31;
  int half = lane >> 4, lr = lane & 15;
  int n0 = wave * 16;
  v8f acc = Z8;
  const v2f* ap = (const v2f*)(u + (size_t)(m0 + lr) * 128);
  const v2f* bp = (const v2f*)(out_w + (size_t)(n0 + lr) * 128);
  for (int kb = 0; kb < 32; ++kb) {
    acc = wmma4v(ap[kb * 2 + half], bp[kb * 2 + half], acc);
  }
  for (int r = 0; r < 8; ++r)
    mbuf[(size_t)(m0 + r + half * 8) * 64 + n0 + lr] = acc[r];
}

// K8: layernorm(m) + residual h, transpose, maxpool2 -> x1p (b,64,1024)
__global__ void k_ln_res_pool(const float* __restrict__ mbuf, const float* __restrict__ hbuf,
                              const float* __restrict__ g, const float* __restrict__ beta,
                              float* __restrict__ x1p) {
  __shared__ float red[64];
  int tp = blockIdx.x;   // 0..1023
  int b = blockIdx.y;
  int d = threadIdx.x;
  float vals[2];
  for (int q = 0; q < 2; ++q) {
    int t = tp * 2 + q;
    size_t tok = (size_t)b * 2048 + t;
    float v = mbuf[tok * 64 + d];
    red[d] = v; __syncthreads();
    for (int s = 32; s > 0; s >>= 1) { if (d < s) red[d] += red[d + s]; __syncthreads(); }
    float mu = red[0] * (1.f / 64.f); __syncthreads();
    float dv = v - mu;
    red[d] = dv * dv; __syncthreads();
    for (int s = 32; s > 0; s >>= 1) { if (d < s) red[d] += red[d + s]; __syncthreads(); }
    float var = red[0] * (1.f / 64.f); __syncthreads();
    vals[q] = dv * rsqrtf(var + 1e-5f) * g[d] + beta[d] + hbuf[tok * 64 + d];
  }
  x1p[((size_t)b * 64 + d) * 1024 + tp] = fmaxf(vals[0], vals[1]);
}

// K9: enc2 conv 64->128 k3 pad1 over L=1024, maxpool2 fused in WMMA store
__global__ void k_enc2(const float* __restrict__ x1p, const float* __restrict__ wt,
                       const float* __restrict__ bias, float* __restrict__ x2p) {
  int mt = blockIdx.x;   // 0..63
  int nb = blockIdx.y;   // 0..1
  int b  = blockIdx.z;
  int wave = threadIdx.x >> 5, lane = threadIdx.x & 31;
  int half = lane >> 4, lr = lane & 15;
  int n0 = nb * 64 + wave * 16;
  v8f acc = Z8;
  for (int tap = 0; tap < 3; ++tap) {
    int t = mt * 16 + lr + tap - 1;
    float msk = (t >= 0 && t < 1024) ? 1.f : 0.f;
    int tc = t < 0 ? 0 : (t > 1023 ? 1023 : t);
    const v2f* wp = (const v2f*)(wt + ((size_t)tap * 128 + n0 + lr) * 64);
    for (int kb = 0; kb < 16; ++kb) {
      int kA = kb * 4 + half * 2;
      float ax = x1p[((size_t)b * 64 + kA) * 1024 + tc] * msk;
      float ay = x1p[((size_t)b * 64 + kA + 1) * 1024 + tc] * msk;
      v2f bv = wp[kb * 2 + half];
      acc = wmma4(ax, ay, bv.x, bv.y, acc);
    }
  }
  int o = n0 + lr;
  float bb = bias[o];
  for (int r = 0; r < 4; ++r) {
    int tpI = mt * 8 + r + half * 4;
    x2p[((size_t)b * 128 + o) * 512 + tpI] = fmaxf(acc[2 * r], acc[2 * r + 1]) + bb;
  }
}

// K10: enc3 conv 128->256 k3 pad1 over L=512, maxpool2 fused; writes d_out
__global__ void k_enc3(const float* __restrict__ x2p, const float* __restrict__ wt,
                       const float* __restrict__ bias, float* __restrict__ out) {
  int mt = blockIdx.x;   // 0..31
  int nb = blockIdx.y;   // 0..3
  int b  = blockIdx.z;
  int wave = threadIdx.x >> 5, lane = threadIdx.x & 31;
  int half = lane >> 4, lr = lane & 15;
  int n0 = nb * 64 + wave * 16;
  v8f acc = Z8;
  for (int tap = 0; tap < 3; ++tap) {
    int t = mt * 16 + lr + tap - 1;
    float msk = (t >= 0 && t < 512) ? 1.f : 0.f;
    int tc = t < 0 ? 0 : (t > 511 ? 511 : t);
    const v2f* wp = (const v2f*)(wt + ((size_t)tap * 256 + n0 + lr) * 128);
    for (int kb = 0; kb < 32; ++kb) {
      int kA = kb * 4 + half * 2;
      float ax = x2p[((size_t)b * 128 + kA) * 512 + tc] * msk;
      float ay = x2p[((size_t)b * 128 + kA + 1) * 512 + tc] * msk;
      v2f bv = wp[kb * 2 + half];
      acc = wmma4(ax, ay, bv.x, bv.y, acc);
    }
  }
  int o = n0 + lr;
  float bb = bias[o];
  for (int r = 0; r < 4; ++r) {
    int tpI = mt * 8 + r + half * 4;
    out[((size_t)b * 256 + o) * 256 + tpI] = fmaxf(acc[2 * r], acc[2 * r + 1]) + bb;
  }
}

// ---------------------------------------------------------------------------
extern "C" void kernel_launch(void* const* d_in, const int* in_sizes, int n_in,
                              void* d_out, int out_size, void* d_ws, size_t ws_size,
                              hipStream_t stream) {
  const float* x       = (const float*)d_in[0];
  const float* emb_w1  = (const float*)d_in[1];
  const float* emb_b1  = (const float*)d_in[2];
  const float* emb_w3  = (const float*)d_in[3];
  const float* emb_b3  = (const float*)d_in[4];
  const float* emb_w7  = (const float*)d_in[5];
  const float* emb_b7  = (const float*)d_in[6];
  const float* fuse_w  = (const float*)d_in[7];
  const float* fuse_b  = (const float*)d_in[8];
  const float* lin_w   = (const float*)d_in[9];
  const float* lin_b   = (const float*)d_in[10];
  const float* in_w    = (const float*)d_in[11];
  const float* conv_w  = (const float*)d_in[12];
  const float* conv_b  = (const float*)d_in[13];
  const float* dt_bias = (const float*)d_in[14];
  const float* A_log   = (const float*)d_in[15];
  const float* D_param = (const float*)d_in[16];
  const float* norm_w  = (const float*)d_in[17];
  const float* out_w   = (const float*)d_in[18];
  const float* ln_g    = (const float*)d_in[19];
  const float* ln_b    = (const float*)d_in[20];
  const float* enc2_w  = (const float*)d_in[21];
  const float* enc2_b  = (const float*)d_in[22];
  const float* enc3_w  = (const float*)d_in[23];
  const float* enc3_b  = (const float*)d_in[24];
  float* out = (float*)d_out;
  float* ws  = (float*)d_ws;

  // workspace layout (floats)
  const size_t o_h    = 0;                       // 2,097,152
  const size_t o_z    = o_h    + 2097152;        // 4,194,304
  const size_t o_xBCr = o_z    + 4194304;        // 8,388,608
  const size_t o_xBCa = o_xBCr + 8388608;        // 8,388,608
  const size_t o_xsd  = o_xBCa + 8388608;        // 4,194,304
  const size_t o_dt   = o_xsd  + 4194304;        // 65,536
  const size_t o_Acs  = o_dt   + 65536;          // 65,536
  const size_t o_G    = o_Acs  + 65536;          // 65,536
  const size_t o_ds   = o_G    + 65536;          // 65,536
  const size_t o_AL   = o_ds   + 65536;          // 256
  const size_t o_S    = o_AL   + 256;            // 524,288
  const size_t o_Sin  = o_S    + 524288;         // 524,288
  const size_t o_y    = o_Sin  + 524288;         // 4,194,304
  const size_t o_m    = o_y    + 4194304;        // 2,097,152
  const size_t o_x1p  = o_m    + 2097152;        // 1,048,576
  const size_t o_x2p  = o_x1p  + 1048576;        // 1,048,576
  const size_t o_Weff = o_x2p  + 1048576;        // 28,672
  const size_t o_W2   = o_Weff + 28672;          // 28,672 ([tap][d][i])
  const size_t o_b2   = o_W2   + 28672;          // 64
  const size_t o_e2w  = o_b2   + 64;             // 24,576 ([tap][o][i])
  const size_t o_e3w  = o_e2w  + 24576;          // 98,304 ([tap][o][i])

  k_prep_weff<<<112, 256, 0, stream>>>(fuse_w, emb_w1, emb_w3, emb_w7, ws + o_Weff);
  k_prep_w2  <<<112, 256, 0, stream>>>(lin_w, ws + o_Weff, ws + o_W2);
  k_prep_bias<<<1, 64, 0, stream>>>(fuse_w, fuse_b, emb_b1, emb_b3, emb_b7,
                                    lin_w, lin_b, ws + o_b2);
  k_prep_enc2<<<96, 256, 0, stream>>>(enc2_w, ws + o_e2w);
  k_prep_enc3<<<384, 256, 0, stream>>>(enc3_w, ws + o_e3w);

  k_embed<<<dim3(128, 16), 128, 0, stream>>>(x, ws + o_W2, ws + o_b2, ws + o_h);

  k_inproj<<<dim3(2048, 6), 128, 0, stream>>>(ws + o_h, in_w, ws + o_z, ws + o_xBCr);
  k_dt<<<256, 256, 0, stream>>>(ws + o_h, in_w, dt_bias, ws + o_dt);

  k_conv_silu<<<32768, 256, 0, stream>>>(ws + o_xBCr, conv_w, conv_b, ws + o_dt,
                                         ws + o_xBCa, ws + o_xsd);

  k_scan<<<1, 256, 0, stream>>>(ws + o_dt, A_log, ws + o_Acs, ws + o_G,
                                ws + o_ds, ws + o_AL);

  k_states<<<256, 128, 0, stream>>>(ws + o_xBCa, ws + o_xsd, ws + o_ds, ws + o_S);
  k_chunkscan<<<32, 256, 0, stream>>>(ws + o_S, ws + o_AL, ws + o_Sin);

  k_ssd_y<<<dim3(8, 16, 16), 256, 0, stream>>>(ws + o_xBCa, ws + o_xsd, ws + o_Acs,
                                               ws + o_G, ws + o_Sin, D_param, ws + o_y);

  k_gate_rms<<<32768, 128, 0, stream>>>(ws + o_y, ws + o_z, norm_w);
  k_outproj<<<2048, 128, 0, stream>>>(ws + o_z, out_w, ws + o_m);

  k_ln_res_pool<<<dim3(1024, 16), 64, 0, stream>>>(ws + o_m, ws + o_h, ln_g, ln_b,
                                                   ws + o_x1p);

  k_enc2<<<dim3(64, 2, 16), 128, 0, stream>>>(ws + o_x1p, ws + o_e2w, enc2_b, ws + o_x2p);
  k_enc3<<<dim3(32, 4, 16), 128, 0, stream>>>(ws + o_x2p, ws + o_e3w, enc3_b, out);

  (void)in_sizes; (void)n_in; (void)out_size; (void)ws_size;
}